// EdgeConv_43499428774200
// MI455X (gfx1250) — compile-verified
//
#include <hip/hip_runtime.h>

// ---------------------------------------------------------------------------
// EdgeConv fused pipeline for MI455X (gfx1250, wave32, WMMA).
// B=8, C_IN=64, C_OUT=128, N=2048, K=16.
// fp32 end-to-end using V_WMMA_F32_16X16X4_F32 (exact fp32 products, fp32 acc)
// ---------------------------------------------------------------------------

typedef __attribute__((ext_vector_type(2))) float v2f;
typedef __attribute__((ext_vector_type(8))) float v8f;

#define BATCH 8
#define CIN   64
#define COUT  128
#define NPTS  2048
#define KNN   16
#define EPSBN 1e-5f
#define NEGINF (-3.0e38f)

static __device__ __forceinline__ v8f wmma_f32x4(v2f a, v2f b, v8f c) {
  // 8 args: (neg_a, A, neg_b, B, c_mod, C, reuse_a, reuse_b)
  return __builtin_amdgcn_wmma_f32_16x16x4_f32(
      false, a, false, b, (short)0, c, false, false);
}

// ---------------------------------------------------------------------------
// Kernel 1: sq[b,n] = sum_c x[b,c,n]^2
// ---------------------------------------------------------------------------
__global__ __launch_bounds__(256) void sq_kernel(const float* __restrict__ x,
                                                 float* __restrict__ sq) {
  int gid = blockIdx.x * 256 + threadIdx.x;       // in [0, B*N)
  int b = gid / NPTS;
  int n = gid - b * NPTS;
  const float* xb = x + (size_t)b * CIN * NPTS + n;
  float acc = 0.f;
#pragma unroll 8
  for (int c = 0; c < CIN; ++c) {
    float v = xb[(size_t)c * NPTS];
    acc += v * v;
  }
  sq[gid] = acc;
}

// ---------------------------------------------------------------------------
// Kernel 2: zero the BN accumulators (2*COUT floats). Re-run every call so
// graph replay stays deterministic.
// ---------------------------------------------------------------------------
__global__ void zero_sums_kernel(float* __restrict__ sums) {
  sums[threadIdx.x] = 0.f;
}

// ---------------------------------------------------------------------------
// Kernel 3: top-K neighbors.
// One workgroup (256 threads = 8 waves) handles 16 query points of one batch.
// Phase A: 16 x 2048 strip of pd = 2*gram - sq_n - sq_m via f32 WMMA into LDS
//          (16*2048*4 = 128 KB dynamic LDS; WGP has 320 KB).
// Phase B: per row, 16 rounds of wave-wide argmax extraction.
//
// f32 16x16x4 WMMA operand layout (ISA 7.12.2, 32-bit A 16x4):
//   A: lanes 0-15 hold row M=lane, VGPR0/1 = K=0/1; lanes 16-31: K=2/3.
//   B (4x16) mirrored: lanes 0-15 col N=lane, K=0/1; lanes 16-31: K=2/3.
//   C: VGPR r, lanes 0-15 -> row r, lanes 16-31 -> row r+8; col = lane&15.
// ---------------------------------------------------------------------------
__global__ __launch_bounds__(256) void topk_kernel(const float* __restrict__ x,
                                                   const float* __restrict__ sq,
                                                   int* __restrict__ knn) {
  extern __shared__ float pd[];                    // 16 * NPTS floats
  const int blk  = blockIdx.x;                     // B * (N/16)
  const int b    = blk / (NPTS / 16);
  const int n0   = (blk - b * (NPTS / 16)) * 16;
  const int tid  = threadIdx.x;
  const int wave = tid >> 5;
  const int lane = tid & 31;
  const int half = lane >> 4;
  const int l15  = lane & 15;

  const float* xb  = x + (size_t)b * CIN * NPTS;
  const float* sqb = sq + b * NPTS;

  // Preload A fragments for the 16 query points: 16 k-steps over CIN=64.
  v2f afrag[16];
#pragma unroll
  for (int s = 0; s < 16; ++s) {
    int c = 4 * s + 2 * half;
    afrag[s].x = xb[(size_t)c * NPTS + n0 + l15];
    afrag[s].y = xb[(size_t)(c + 1) * NPTS + n0 + l15];
  }
  // Per-row sq of the 16 queries this block owns (eight rows matter per lane).
  float sqn_lo = sqb[n0 + (0 + 8 * half) /*overwritten in loop*/];
  (void)sqn_lo;

  // Phase A: each wave covers 16 of the 128 m-tiles.
  for (int tt = 0; tt < 16; ++tt) {
    const int m0 = (wave + 8 * tt) * 16;
    v8f acc = {0.f, 0.f, 0.f, 0.f, 0.f, 0.f, 0.f, 0.f};
#pragma unroll
    for (int s = 0; s < 16; ++s) {
      int c = 4 * s + 2 * half;
      v2f bfrag;
      bfrag.x = xb[(size_t)c * NPTS + m0 + l15];
      bfrag.y = xb[(size_t)(c + 1) * NPTS + m0 + l15];
      acc = wmma_f32x4(afrag[s], bfrag, acc);
    }
    const float sqm = sqb[m0 + l15];
#pragma unroll
    for (int r = 0; r < 8; ++r) {
      const int row = r + 8 * half;
      const float v = 2.f * acc[r] - sqb[n0 + row] - sqm;
      pd[row * NPTS + m0 + l15] = v;
    }
  }
  __syncthreads();

  // Phase B: wave w extracts top-16 for rows 2w and 2w+1.
  for (int rr = 0; rr < 2; ++rr) {
    const int row = 2 * wave + rr;
    volatile float* prow = pd + row * NPTS;
    for (int it = 0; it < KNN; ++it) {
      float bestv = NEGINF;
      int   besti = NPTS;
      for (int m = lane; m < NPTS; m += 32) {
        float v = prow[m];
        if (v > bestv || (v == bestv && m < besti)) { bestv = v; besti = m; }
      }
#pragma unroll
      for (int off = 16; off > 0; off >>= 1) {
        float ov = __shfl_xor(bestv, off, 32);
        int   oi = __shfl_xor(besti, off, 32);
        if (ov > bestv || (ov == bestv && oi < besti)) { bestv = ov; besti = oi; }
      }
      if (lane == 0) {
        knn[((size_t)b * NPTS + n0 + row) * KNN + it] = besti;
        prow[besti] = NEGINF;   // wave-private row; DS ops in-order per wave
      }
    }
  }
}

// ---------------------------------------------------------------------------
// Kernel 4: edge GEMM + ReLU + fused BN-stat / max / min over K.
// One wave per query point: A = e (16 edges x 128 feat), B = W^T tile from LDS.
// h = relu(e @ W^T + b). Per output channel we keep max_k, min_k and add
// (sum, sumsq) into global accumulators. h itself is never stored.
// ---------------------------------------------------------------------------
__global__ __launch_bounds__(256) void edge_kernel(
    const float* __restrict__ x, const float* __restrict__ W,
    const float* __restrict__ bias, const int* __restrict__ knn,
    float* __restrict__ hmax, float* __restrict__ hmin,
    float* __restrict__ sums) {
  extern __shared__ float smem[];
  float* Wl   = smem;                      // COUT * 2*CIN = 16384
  float* bl   = Wl + COUT * 2 * CIN;       // COUT
  float* ssum = bl + COUT;                 // COUT
  float* ssq  = ssum + COUT;               // COUT

  const int tid = threadIdx.x;
  __builtin_prefetch(W + tid * 4, 0, 0);   // global_prefetch_b8 warm-up
  for (int i = tid; i < COUT * 2 * CIN; i += 256) Wl[i] = W[i];
  if (tid < COUT) { bl[tid] = bias[tid]; ssum[tid] = 0.f; ssq[tid] = 0.f; }
  __syncthreads();

  const int wave = tid >> 5;
  const int lane = tid & 31;
  const int half = lane >> 4;
  const int l15  = lane & 15;

  const size_t q = (size_t)blockIdx.x * 8 + wave;   // query id in [0, B*N)
  const int b = (int)(q / NPTS);
  const int n = (int)(q - (size_t)b * NPTS);
  const float* xb = x + (size_t)b * CIN * NPTS;
  const int j = knn[q * KNN + l15];                 // lane's edge row k = l15

  // Build A fragments: 32 k-steps over the 128 edge features.
  // c < 64  -> x_i[c]           (broadcast across edge rows)
  // c >= 64 -> x_j[c-64]-x_i[c-64] (per-row gather)
  v2f afrag[32];
#pragma unroll
  for (int s = 0; s < 32; ++s) {
    const int c0 = 4 * s + 2 * half;   // c0 and c0+1 stay on one side of 64
    float a0, a1;
    if (c0 < CIN) {
      a0 = xb[(size_t)c0 * NPTS + n];
      a1 = xb[(size_t)(c0 + 1) * NPTS + n];
    } else {
      const int cj = c0 - CIN;
      a0 = xb[(size_t)cj * NPTS + j] - xb[(size_t)cj * NPTS + n];
      a1 = xb[(size_t)(cj + 1) * NPTS + j] - xb[(size_t)(cj + 1) * NPTS + n];
    }
    afrag[s].x = a0;
    afrag[s].y = a1;
  }

#pragma unroll 1
  for (int ot = 0; ot < 8; ++ot) {
    const int o0  = ot * 16;
    const int col = o0 + l15;
    v8f acc = {0.f, 0.f, 0.f, 0.f, 0.f, 0.f, 0.f, 0.f};
#pragma unroll
    for (int s = 0; s < 32; ++s) {
      const int c0 = 4 * s + 2 * half;
      v2f bfrag;                                   // B(K=c, N=col) = W[col][c]
      bfrag.x = Wl[col * (2 * CIN) + c0];
      bfrag.y = Wl[col * (2 * CIN) + c0 + 1];
      acc = wmma_f32x4(afrag[s], bfrag, acc);
    }
    const float bv = bl[col];
    float mx = NEGINF, mn = -NEGINF, sm = 0.f, s2 = 0.f;
#pragma unroll
    for (int r = 0; r < 8; ++r) {                  // rows k = r + 8*half
      float h = acc[r] + bv;
      h = h > 0.f ? h : 0.f;                       // ReLU
      mx = fmaxf(mx, h);
      mn = fminf(mn, h);
      sm += h;
      s2 += h * h;
    }
    // lanes l and l^16 hold the same channel, complementary k-rows
    mx = fmaxf(mx, __shfl_xor(mx, 16, 32));
    mn = fminf(mn, __shfl_xor(mn, 16, 32));
    sm += __shfl_xor(sm, 16, 32);
    s2 += __shfl_xor(s2, 16, 32);
    if (half == 0) {
      hmax[q * COUT + col] = mx;
      hmin[q * COUT + col] = mn;
      atomicAdd(&ssum[col], sm);
      atomicAdd(&ssq[col], s2);
    }
  }
  __syncthreads();
  if (tid < COUT) {
    atomicAdd(&sums[tid], ssum[tid]);
    atomicAdd(&sums[COUT + tid], ssq[tid]);
  }
}

// ---------------------------------------------------------------------------
// Kernel 5: finalize. BN is per-channel affine; max over K commutes with a
// positive-slope affine and flips to min for negative slope.
// out layout: (B, C_OUT, N, 1).
// ---------------------------------------------------------------------------
__global__ __launch_bounds__(256) void finalize_kernel(
    const float* __restrict__ hmax, const float* __restrict__ hmin,
    const float* __restrict__ sums, const float* __restrict__ gamma,
    const float* __restrict__ beta, float* __restrict__ out) {
  const size_t gid = (size_t)blockIdx.x * 256 + threadIdx.x;  // B*COUT*N
  const int n = (int)(gid % NPTS);
  const int c = (int)((gid / NPTS) % COUT);
  const int b = (int)(gid / ((size_t)NPTS * COUT));

  const float cnt  = (float)BATCH * (float)NPTS * (float)KNN;
  const float mean = sums[c] / cnt;
  float var = sums[COUT + c] / cnt - mean * mean;
  var = var > 0.f ? var : 0.f;
  const float scale = gamma[c] * rsqrtf(var + EPSBN);
  const float shift = beta[c] - mean * scale;

  const size_t qi = ((size_t)b * NPTS + n) * COUT + c;
  const float v = (scale >= 0.f) ? hmax[qi] * scale + shift
                                 : hmin[qi] * scale + shift;
  out[gid] = v;
}

// ---------------------------------------------------------------------------
// Host-side launcher (graph-capture safe: only kernel launches on `stream`).
// ---------------------------------------------------------------------------
extern "C" void kernel_launch(void* const* d_in, const int* in_sizes, int n_in,
                              void* d_out, int out_size, void* d_ws,
                              size_t ws_size, hipStream_t stream) {
  (void)in_sizes; (void)n_in; (void)out_size; (void)ws_size;
  const float* x     = (const float*)d_in[0];   // (B, C_IN, N, 1)
  const float* W     = (const float*)d_in[1];   // (C_OUT, 2*C_IN)
  const float* bias  = (const float*)d_in[2];   // (C_OUT)
  const float* gamma = (const float*)d_in[3];   // (C_OUT)
  const float* beta  = (const float*)d_in[4];   // (C_OUT)
  float* out = (float*)d_out;                   // (B, C_OUT, N, 1)

  // Workspace layout (~17.9 MB total):
  char* ws = (char*)d_ws;
  float* sq   = (float*)ws;                                   // B*N floats
  int*   knn  = (int*)(ws + (size_t)BATCH * NPTS * 4);        // B*N*K ints
  float* hmax = (float*)(ws + (size_t)BATCH * NPTS * 4
                            + (size_t)BATCH * NPTS * KNN * 4);
  float* hmin = hmax + (size_t)BATCH * NPTS * COUT;
  float* sums = hmin + (size_t)BATCH * NPTS * COUT;           // 2*COUT floats

  sq_kernel<<<(BATCH * NPTS) / 256, 256, 0, stream>>>(x, sq);
  zero_sums_kernel<<<1, 2 * COUT, 0, stream>>>(sums);

  const size_t topk_lds = (size_t)16 * NPTS * sizeof(float);  // 128 KB
  topk_kernel<<<BATCH * (NPTS / 16), 256, topk_lds, stream>>>(x, sq, knn);

  const size_t edge_lds = (size_t)(COUT * 2 * CIN + 3 * COUT) * sizeof(float);
  edge_kernel<<<(BATCH * NPTS) / 8, 256, edge_lds, stream>>>(x, W, bias, knn,
                                                             hmax, hmin, sums);

  finalize_kernel<<<(BATCH * COUT * NPTS) / 256, 256, 0, stream>>>(
      hmax, hmin, sums, gamma, beta, out);
}